// LoRARouter_42528766165292
// MI455X (gfx1250) — compile-verified
//
#include <hip/hip_runtime.h>

typedef __attribute__((ext_vector_type(2))) float v2f;
typedef __attribute__((ext_vector_type(8))) float v8f;

static constexpr int kD = 1024;   // d_model
static constexpr int kB = 32768;  // batch
static constexpr int kM = 4;      // modules

// ---------------------------------------------------------------------------
// Kernel 1: Wcomb[m][d] = sum_e Wr[m][e] * Wg[e][d]   ([4,1024] = [4,1024]x[1024,1024])
// Tiny (reads 4 MB); plain VALU FMA, Wr broadcast from LDS.
// ---------------------------------------------------------------------------
__global__ void __launch_bounds__(256)
wcomb_kernel(const float* __restrict__ Wg, const float* __restrict__ Wr,
             float* __restrict__ Wcomb) {
  __shared__ float sWr[kM * kD];
  const int tid = threadIdx.x;
  for (int i = tid; i < kM * kD; i += 256) sWr[i] = Wr[i];
  __syncthreads();

  const int d = blockIdx.x * 256 + tid;  // 0..1023 over 4 blocks
  float a0 = 0.f, a1 = 0.f, a2 = 0.f, a3 = 0.f;
#pragma unroll 8
  for (int e = 0; e < kD; ++e) {
    const float g = Wg[(size_t)e * kD + d];  // coalesced across threads
    a0 = fmaf(sWr[0 * kD + e], g, a0);
    a1 = fmaf(sWr[1 * kD + e], g, a1);
    a2 = fmaf(sWr[2 * kD + e], g, a2);
    a3 = fmaf(sWr[3 * kD + e], g, a3);
  }
  Wcomb[0 * kD + d] = a0;
  Wcomb[1 * kD + d] = a1;
  Wcomb[2 * kD + d] = a2;
  Wcomb[3 * kD + d] = a3;
}

// ---------------------------------------------------------------------------
// Kernel 2: logits = h @ Wcomb^T via V_WMMA_F32_16X16X4_F32, then softmax(4)
// threshold -> constant weight vectors, written as [M][B][4].
// One wave32 per 16-row tile; 8 waves / 256 threads per block -> 128 rows/blk.
// ---------------------------------------------------------------------------
__global__ void __launch_bounds__(256)
router_kernel(const float* __restrict__ h, const float* __restrict__ Wcomb,
              float* __restrict__ out) {
  // Rows 0-3: Wcomb (B-matrix, N=module). Row 4: zeros (pads N=4..15 of WMMA).
  __shared__ float sW[5 * kD];     // 20 KB
  __shared__ float sLog[8][16][4]; // per-wave logits scratch, 2 KB

  const int tid = threadIdx.x;
  for (int i = tid; i < 5 * kD; i += 256)
    sW[i] = (i < kM * kD) ? Wcomb[i] : 0.0f;
  __syncthreads();

  const int wave    = tid >> 5;   // 0..7
  const int lane    = tid & 31;
  const int lane16  = lane & 15;
  const int halfSel = lane >> 4;  // 0: K+{0,1}, 1: K+{2,3}

  const int rowTile = blockIdx.x * 128 + wave * 16;

  // A frag source: lane16 = row within tile, halfSel selects K pair.
  const float* aptr = h + (size_t)(rowTile + lane16) * kD + halfSel * 2;
  // B frag source: N = lane16 (module); N>=4 reads the zero row.
  const int nrow = (lane16 < kM) ? lane16 : kM;
  const float* bptr = sW + nrow * kD + halfSel * 2;

  v8f c0 = {};
  v8f c1 = {};
#pragma unroll 4
  for (int k = 0; k < kD; k += 8) {
    v2f a0 = *(const v2f*)(aptr + k);
    v2f b0 = *(const v2f*)(bptr + k);
    v2f a1 = *(const v2f*)(aptr + k + 4);
    v2f b1 = *(const v2f*)(bptr + k + 4);
    c0 = __builtin_amdgcn_wmma_f32_16x16x4_f32(false, a0, false, b0,
                                               (short)0, c0, false, false);
    c1 = __builtin_amdgcn_wmma_f32_16x16x4_f32(false, a1, false, b1,
                                               (short)0, c1, false, false);
  }
  v8f c = c0 + c1;

  // C/D layout: VGPR i -> M = i + halfSel*8, N = lane16. Park logits in LDS.
  if (lane16 < kM) {
#pragma unroll
    for (int i = 0; i < 8; ++i)
      sLog[wave][i + halfSel * 8][lane16] = c[i];
  }
  __syncthreads();

  if (lane < 16) {
    const int b = rowTile + lane;
    const float l0 = sLog[wave][lane][0];
    const float l1 = sLog[wave][lane][1];
    const float l2 = sLog[wave][lane][2];
    const float l3 = sLog[wave][lane][3];
    const float mx = fmaxf(fmaxf(l0, l1), fmaxf(l2, l3));
    const float e0 = __expf(l0 - mx);
    const float e1 = __expf(l1 - mx);
    const float e2 = __expf(l2 - mx);
    const float e3 = __expf(l3 - mx);
    const float hs = 0.5f * (e0 + e1 + e2 + e3);  // prob_m > 0.5  <=>  e_m > hs

    const float4 w_hi = make_float4(0.5f, 0.5f, 0.0f, 0.0f);  // TOP_K=2
    const float4 w_lo = make_float4(1.0f, 0.0f, 0.0f, 0.0f);  // n_lo=1

    float4* o = reinterpret_cast<float4*>(out);
    o[(size_t)0 * kB + b] = (e0 > hs) ? w_hi : w_lo;
    o[(size_t)1 * kB + b] = (e1 > hs) ? w_hi : w_lo;
    o[(size_t)2 * kB + b] = (e2 > hs) ? w_hi : w_lo;
    o[(size_t)3 * kB + b] = (e3 > hs) ? w_hi : w_lo;
  }
}

// ---------------------------------------------------------------------------
extern "C" void kernel_launch(void* const* d_in, const int* in_sizes, int n_in,
                              void* d_out, int out_size, void* d_ws,
                              size_t ws_size, hipStream_t stream) {
  const float* h  = (const float*)d_in[0];  // [32768,1024]
  const float* Wg = (const float*)d_in[1];  // [1024,1024]
  const float* Wr = (const float*)d_in[2];  // [4,1024]
  float* Wcomb = (float*)d_ws;              // 16 KB scratch
  float* out   = (float*)d_out;             // [4][32768][4]

  wcomb_kernel<<<kD / 256, 256, 0, stream>>>(Wg, Wr, Wcomb);
  router_kernel<<<kB / 128, 256, 0, stream>>>(h, Wcomb, out);
}